// SelfAttentionBlock_15908558864510
// MI455X (gfx1250) — compile-verified
//
#include <hip/hip_runtime.h>
#include <hip/hip_bf16.h>

typedef __attribute__((ext_vector_type(2))) float v2f;
typedef __attribute__((ext_vector_type(4))) float v4f;
typedef __attribute__((ext_vector_type(8))) float v8f;

#define N_EMBED 1024
#define HEAD    64
#define MROWS   8192   // B*T = 4*2048

// out[m, n] = sum_k x[m,k] * Wv[n,k] + bv[n]
// (entire reference reduces to the V projection: softmax rows sum to 1,
//  and einsum('btu,btj->btj') multiplies that row-sum elementwise with v)
//
// K remap trick: matrix-K order inside a WMMA is irrelevant to the reduction,
// so each lane loads a 16B-aligned b128 covering 4 consecutive k's:
//   lanes 0-15  : mem k0+0..3      lanes 16-31 : mem k0+4..7
// WMMA step 1 consumes (.x,.y) of A and B -> mem k {0,1,4,5}
// WMMA step 2 consumes (.z,.w)            -> mem k {2,3,6,7}
// A and B use the same mapping, so the sum over k is exact.
__global__ __launch_bounds__(256) void vproj_wmma_f32(
    const float* __restrict__ x,    // [MROWS, N_EMBED]
    const float* __restrict__ Wv,   // [HEAD, N_EMBED]
    const float* __restrict__ bv,   // [HEAD]
    float* __restrict__ out)        // [MROWS, HEAD]
{
    const int lane = threadIdx.x & 31;
    const int wave = threadIdx.x >> 5;
    const int m0   = (blockIdx.x * 8 + wave) * 16;   // 16-row block per wave
    const int half = lane >> 4;                      // 0 or 1
    const int l15  = lane & 15;
    const int kh   = half << 2;                      // K sub-offset: 0 or 4

    // --- Seed accumulators with bias: C[r] for this lane is out[*, nt*16 + l15]
    v8f acc[4];
#pragma unroll
    for (int nt = 0; nt < 4; ++nt) {
        const float b = bv[nt * 16 + l15];
#pragma unroll
        for (int r = 0; r < 8; ++r) acc[nt][r] = b;
    }

    // A fragment source: row m0+l15 of x; lane halves split across K (see above)
    const float* __restrict__ xrow = x + (size_t)(m0 + l15) * N_EMBED + kh;

    // B fragment sources: B[k][n] = Wv[n][k]; lane's column n = nt*16 + l15
    const float* __restrict__ wrow0 = Wv + (size_t)(0  + l15) * N_EMBED + kh;
    const float* __restrict__ wrow1 = Wv + (size_t)(16 + l15) * N_EMBED + kh;
    const float* __restrict__ wrow2 = Wv + (size_t)(32 + l15) * N_EMBED + kh;
    const float* __restrict__ wrow3 = Wv + (size_t)(48 + l15) * N_EMBED + kh;

#pragma unroll 2
    for (int k0 = 0; k0 < N_EMBED; k0 += 8) {
        const v4f a  = *(const v4f*)(xrow  + k0);
        const v4f b0 = *(const v4f*)(wrow0 + k0);
        const v4f b1 = *(const v4f*)(wrow1 + k0);
        const v4f b2 = *(const v4f*)(wrow2 + k0);
        const v4f b3 = *(const v4f*)(wrow3 + k0);

        const v2f a01 = {a.x, a.y};
        const v2f a23 = {a.z, a.w};

        // step 1: mem-k {k0, k0+1, k0+4, k0+5}
        acc[0] = __builtin_amdgcn_wmma_f32_16x16x4_f32(false, a01, false, (v2f){b0.x, b0.y},
                                                       (short)0, acc[0], false, false);
        acc[1] = __builtin_amdgcn_wmma_f32_16x16x4_f32(false, a01, false, (v2f){b1.x, b1.y},
                                                       (short)0, acc[1], false, false);
        acc[2] = __builtin_amdgcn_wmma_f32_16x16x4_f32(false, a01, false, (v2f){b2.x, b2.y},
                                                       (short)0, acc[2], false, false);
        acc[3] = __builtin_amdgcn_wmma_f32_16x16x4_f32(false, a01, false, (v2f){b3.x, b3.y},
                                                       (short)0, acc[3], false, false);
        // step 2: mem-k {k0+2, k0+3, k0+6, k0+7}
        acc[0] = __builtin_amdgcn_wmma_f32_16x16x4_f32(false, a23, false, (v2f){b0.z, b0.w},
                                                       (short)0, acc[0], false, false);
        acc[1] = __builtin_amdgcn_wmma_f32_16x16x4_f32(false, a23, false, (v2f){b1.z, b1.w},
                                                       (short)0, acc[1], false, false);
        acc[2] = __builtin_amdgcn_wmma_f32_16x16x4_f32(false, a23, false, (v2f){b2.z, b2.w},
                                                       (short)0, acc[2], false, false);
        acc[3] = __builtin_amdgcn_wmma_f32_16x16x4_f32(false, a23, false, (v2f){b3.z, b3.w},
                                                       (short)0, acc[3], false, false);
    }

    // --- Store: D tile layout -> VGPR r holds rows m0+r (lanes 0-15) and
    // m0+8+r (lanes 16-31), column nt*16 + l15.
    const int mr_base = m0 + (half << 3);
#pragma unroll
    for (int nt = 0; nt < 4; ++nt) {
        float* __restrict__ o = out + (size_t)mr_base * HEAD + nt * 16 + l15;
#pragma unroll
        for (int r = 0; r < 8; ++r) {
            o[(size_t)r * HEAD] = acc[nt][r];
        }
    }
}

extern "C" void kernel_launch(void* const* d_in, const int* in_sizes, int n_in,
                              void* d_out, int out_size, void* d_ws, size_t ws_size,
                              hipStream_t stream) {
    // setup_inputs order: x, Wq, bq, Wk, bk, Wv, bv
    const float* x  = (const float*)d_in[0];
    const float* Wv = (const float*)d_in[5];
    const float* bv = (const float*)d_in[6];
    float* out = (float*)d_out;

    // 8 waves/block * 16 rows/wave = 128 rows per block; 8192 rows total
    dim3 grid(MROWS / 128);
    dim3 block(256);
    vproj_wmma_f32<<<grid, block, 0, stream>>>(x, Wv, bv, out);
}